// SelfAttention_Module_multiscale_v2_14499809591919
// MI455X (gfx1250) — compile-verified
//
#include <hip/hip_runtime.h>
#include <hip/hip_bf16.h>
#include <cstddef>

// ---------------- problem constants ----------------
#define BSZ   4
#define CIN   2048
#define HH    32
#define WW    32
#define NPIX  1024          // H*W
#define DCH   1024          // conv out channels
#define NCLS  21
#define NSC   7
#define KCONV (CIN*9)       // 18432
#define THRESH 0.8f

#if __has_builtin(__builtin_amdgcn_global_load_async_to_lds_b128) && \
    __has_builtin(__builtin_amdgcn_s_wait_asynccnt)
#define HAVE_ASYNC_LDS 1
#else
#define HAVE_ASYNC_LDS 0
#endif

typedef float v2f __attribute__((ext_vector_type(2)));
typedef float v8f __attribute__((ext_vector_type(8)));
typedef int   v4i __attribute__((ext_vector_type(4)));

#define GLOBAL_AS __attribute__((address_space(1)))
#define LDS_AS    __attribute__((address_space(3)))

static __device__ __forceinline__ v8f wmma4(v2f a, v2f b, v8f c) {
  // D = A(16x4,f32) * B(4x16,f32) + C(16x16,f32)   -> v_wmma_f32_16x16x4_f32
  return __builtin_amdgcn_wmma_f32_16x16x4_f32(
      /*neg_a=*/false, a, /*neg_b=*/false, b,
      /*c_mod=*/(short)0, c, /*reuse_a=*/false, /*reuse_b=*/false);
}

// =====================================================================
// K1: dilated 3x3 conv as implicit GEMM.
//   C[d][p] = sum_k conv_w[d][k] * im2col(x)[p][k],  k=(ci,kh,kw)
//   block tile 128(d) x 128(pix), Kt=16, 8 waves (2x4), wave tile 64x32
// =====================================================================
__global__ __launch_bounds__(256)
void conv_gemm_kernel(const float* __restrict__ x,
                      const float* __restrict__ cw,
                      const float* __restrict__ cb,
                      float* __restrict__ feat) {
  __shared__ float As[16][144];   // [k][d]
  __shared__ float Bs[16][144];   // [k][p]
  const int t    = threadIdx.x;
  const int lane = t & 31;
  const int wid  = t >> 5;
  const int wm   = wid >> 2;      // 0..1
  const int wn   = wid & 3;       // 0..3
  const int dT0  = blockIdx.y * 128;
  const int pT0  = blockIdx.x * 128;

  v8f acc[4][2];
  for (int i = 0; i < 4; ++i)
    for (int j = 0; j < 2; ++j) acc[i][j] = (v8f)0.0f;

  for (int kt = 0; kt < KCONV; kt += 16) {
    // ---- stage A: conv_w rows, float4 along k ----
    for (int i = 0; i < 2; ++i) {
      int f4 = i * 256 + t;
      int dl = f4 >> 2;
      int kg = (f4 & 3) << 2;
      const float4 v = *(const float4*)(cw + (size_t)(dT0 + dl) * KCONV + kt + kg);
      As[kg + 0][dl] = v.x; As[kg + 1][dl] = v.y;
      As[kg + 2][dl] = v.z; As[kg + 3][dl] = v.w;
    }
    // ---- stage B: im2col gather (predicated zero padding) ----
    for (int j = 0; j < 8; ++j) {
      int lin = j * 256 + t;
      int k  = lin >> 7;
      int p  = lin & 127;
      int P  = pT0 + p;
      int b  = P >> 10;
      int m  = P & 1023;
      int oh = m >> 5, ow = m & 31;
      int kk = kt + k;
      int ci = kk / 9;
      int r  = kk - ci * 9;
      int kh = r / 3;
      int kw = r - kh * 3;
      int ih = oh + 6 * kh - 6;
      int iw = ow + 6 * kw - 6;
      float v = 0.0f;
      if ((unsigned)ih < 32u && (unsigned)iw < 32u)
        v = x[(((size_t)b * CIN + ci) << 10) + (ih << 5) + iw];
      Bs[k][p] = v;
    }
    __syncthreads();
    if (kt + 16 < KCONV)
      __builtin_prefetch(cw + (size_t)(dT0 + (t >> 1)) * KCONV + kt + 16, 0, 1);

    for (int ks = 0; ks < 4; ++ks) {
      const int kb = ks * 4 + ((lane >> 4) << 1);
      v2f bf[2], af[4];
      for (int ni = 0; ni < 2; ++ni) {
        int nn = wn * 32 + ni * 16 + (lane & 15);
        bf[ni].x = Bs[kb][nn]; bf[ni].y = Bs[kb + 1][nn];
      }
      for (int mi = 0; mi < 4; ++mi) {
        int mm = wm * 64 + mi * 16 + (lane & 15);
        af[mi].x = As[kb][mm]; af[mi].y = As[kb + 1][mm];
      }
      for (int mi = 0; mi < 4; ++mi)
        for (int ni = 0; ni < 2; ++ni)
          acc[mi][ni] = wmma4(af[mi], bf[ni], acc[mi][ni]);
    }
    __syncthreads();
  }
  // ---- epilogue: feat[b][d][m] = acc + conv_b[d] ----
  const int hiM = (lane >> 4) << 3;
  for (int mi = 0; mi < 4; ++mi)
    for (int ni = 0; ni < 2; ++ni)
      for (int r = 0; r < 8; ++r) {
        int d = dT0 + wm * 64 + mi * 16 + r + hiM;
        int P = pT0 + wn * 32 + ni * 16 + (lane & 15);
        int b = P >> 10, m = P & 1023;
        feat[(((size_t)b << 10) + d) * NPIX + m] = acc[mi][ni][r] + cb[d];
      }
}

// =====================================================================
// K2: Dn[s,b,n,m] = thr(diffW[b,n,m]^scale[s]) / rowsum  (diffW layout kept)
// =====================================================================
__global__ __launch_bounds__(256)
void dn_kernel(const float* __restrict__ dw, float* __restrict__ dn) {
  __shared__ float red[256];
  const float SC[NSC] = {0.1f, 1.0f, 5.0f, 10.0f, 20.0f, 50.0f, 100.0f};
  int bid = blockIdx.x;
  int s   = bid / (BSZ * NPIX);
  int rem = bid - s * (BSZ * NPIX);
  int b   = rem >> 10;
  int nrw = rem & 1023;
  const float sc = SC[s];
  const float* row = dw + (((size_t)b << 10) + nrw) * NPIX;
  float pv[4];
  float part = 0.0f;
  for (int j = 0; j < 4; ++j) {
    int m = threadIdx.x + j * 256;
    float w = row[m];
    float p = 0.0f;
    if (w > 0.0f) p = __expf(sc * __logf(w));
    p = (p > THRESH) ? p : 0.0f;
    pv[j] = p;
    part += p;
  }
  red[threadIdx.x] = part;
  __syncthreads();
  for (int off = 128; off > 0; off >>= 1) {
    if (threadIdx.x < off) red[threadIdx.x] += red[threadIdx.x + off];
    __syncthreads();
  }
  float inv = 1.0f / red[0];
  float* orow = dn + ((((size_t)s * BSZ + b) << 10) + nrw) * NPIX;
  for (int j = 0; j < 4; ++j) orow[threadIdx.x + j * 256] = pv[j] * inv;
}

// =====================================================================
// K3: out_diff[b,s,d,n] = sum_m feat[b,d,m] * Dn[s,b,n,m]   (A * B^T)
//   Double-buffered LDS stages; async global->LDS copies when available
//   (GLOBAL_LOAD_ASYNC_TO_LDS_B128 + s_wait_asynccnt), natural [row][k]
//   LDS layout with stride 20 => bank-conflict-free fragment reads and
//   b64-fusable adjacent-k pairs.
// =====================================================================
#define DSTRIDE 20

__global__ __launch_bounds__(256)
void diff_gemm_kernel(const float* __restrict__ feat,
                      const float* __restrict__ dn,
                      float* __restrict__ od) {
  __shared__ float As[2][128][DSTRIDE];   // [buf][d][k]
  __shared__ float Bs[2][128][DSTRIDE];   // [buf][n][k]
  const int t = threadIdx.x, lane = t & 31, wid = t >> 5;
  const int wm = wid >> 2, wn = wid & 3;
  const int z = blockIdx.z;
  const int b = z / NSC, s = z - b * NSC;
  const int dT0 = blockIdx.y * 128;
  const int nT0 = blockIdx.x * 128;
  const float* A = feat + ((size_t)b << 10) * NPIX;              // [d][m]
  const float* B = dn + (((size_t)s * BSZ + b) << 10) * NPIX;    // [n][m]

  v8f acc[4][2];
  for (int i = 0; i < 4; ++i)
    for (int j = 0; j < 2; ++j) acc[i][j] = (v8f)0.0f;

  // per-thread staging coordinates: 2 float4 per matrix per stage
  const int rl0 = t >> 2;                 // row (0..63)
  const int kg  = (t & 3) << 2;           // k group 0/4/8/12

  auto stage = [&](int kt, int buf) {
#if HAVE_ASYNC_LDS
    for (int i = 0; i < 2; ++i) {
      int row = rl0 + i * 64;
      __builtin_amdgcn_global_load_async_to_lds_b128(
          (GLOBAL_AS v4i*)(float*)(A + (size_t)(dT0 + row) * NPIX + kt + kg),
          (LDS_AS v4i*)&As[buf][row][kg],
          0, 0);
      __builtin_amdgcn_global_load_async_to_lds_b128(
          (GLOBAL_AS v4i*)(float*)(B + (size_t)(nT0 + row) * NPIX + kt + kg),
          (LDS_AS v4i*)&Bs[buf][row][kg],
          0, 0);
    }
#else
    for (int i = 0; i < 2; ++i) {
      int row = rl0 + i * 64;
      const float4 va = *(const float4*)(A + (size_t)(dT0 + row) * NPIX + kt + kg);
      As[buf][row][kg + 0] = va.x; As[buf][row][kg + 1] = va.y;
      As[buf][row][kg + 2] = va.z; As[buf][row][kg + 3] = va.w;
      const float4 vb = *(const float4*)(B + (size_t)(nT0 + row) * NPIX + kt + kg);
      Bs[buf][row][kg + 0] = vb.x; Bs[buf][row][kg + 1] = vb.y;
      Bs[buf][row][kg + 2] = vb.z; Bs[buf][row][kg + 3] = vb.w;
    }
#endif
  };

  stage(0, 0);
  int buf = 0;
  for (int kt = 0; kt < NPIX; kt += 16) {
#if HAVE_ASYNC_LDS
    __builtin_amdgcn_s_wait_asynccnt(0);
#endif
    __syncthreads();
    if (kt + 16 < NPIX) stage(kt + 16, buf ^ 1);

    for (int ks = 0; ks < 4; ++ks) {
      const int kb = ks * 4 + ((lane >> 4) << 1);
      v2f bf[2], af[4];
      for (int ni = 0; ni < 2; ++ni) {
        int nn = wn * 32 + ni * 16 + (lane & 15);
        bf[ni].x = Bs[buf][nn][kb]; bf[ni].y = Bs[buf][nn][kb + 1];
      }
      for (int mi = 0; mi < 4; ++mi) {
        int mm = wm * 64 + mi * 16 + (lane & 15);
        af[mi].x = As[buf][mm][kb]; af[mi].y = As[buf][mm][kb + 1];
      }
      for (int mi = 0; mi < 4; ++mi)
        for (int ni = 0; ni < 2; ++ni)
          acc[mi][ni] = wmma4(af[mi], bf[ni], acc[mi][ni]);
    }
    buf ^= 1;
  }
  float* C = od + (((size_t)b * NSC + s) << 10) * NPIX;
  const int hiM = (lane >> 4) << 3;
  for (int mi = 0; mi < 4; ++mi)
    for (int ni = 0; ni < 2; ++ni)
      for (int r = 0; r < 8; ++r) {
        int d = dT0 + wm * 64 + mi * 16 + r + hiM;
        int n = nT0 + wn * 32 + ni * 16 + (lane & 15);
        C[(size_t)d * NPIX + n] = acc[mi][ni][r];
      }
}

// =====================================================================
// K4: wei_raw[b,s,c,n] = sum_d w1[c,d]*out_diff[b,s,d,n] + b1[c]
//     M=32 (21 padded), N tile 128, K=1024, Kt=32
// =====================================================================
__global__ __launch_bounds__(256)
void cls_gemm_kernel(const float* __restrict__ od,
                     const float* __restrict__ w1,
                     const float* __restrict__ b1,
                     float* __restrict__ wr) {
  __shared__ float As[32][48];    // [k=d][c]
  __shared__ float Bs[32][144];   // [k=d][n]
  const int t = threadIdx.x, lane = t & 31, wid = t >> 5;
  const int wm = wid >> 2, wn = wid & 3;
  const int z = blockIdx.z;
  const int b = z / NSC, s = z - b * NSC;
  const int nT0 = blockIdx.x * 128;
  const float* B = od + (((size_t)b * NSC + s) << 10) * NPIX;   // [d][n]

  v8f acc[2];
  acc[0] = (v8f)0.0f; acc[1] = (v8f)0.0f;

  for (int kt = 0; kt < DCH; kt += 32) {
    {
      int cl = t >> 3, kg = (t & 7) << 2;
      float4 v = make_float4(0.f, 0.f, 0.f, 0.f);
      if (cl < NCLS) v = *(const float4*)(w1 + (size_t)cl * DCH + kt + kg);
      As[kg + 0][cl] = v.x; As[kg + 1][cl] = v.y;
      As[kg + 2][cl] = v.z; As[kg + 3][cl] = v.w;
    }
    for (int i = 0; i < 4; ++i) {
      int f4 = i * 256 + t;
      int kl = f4 >> 5, ng = (f4 & 31) << 2;
      const float4 v = *(const float4*)(B + (size_t)(kt + kl) * NPIX + nT0 + ng);
      Bs[kl][ng + 0] = v.x; Bs[kl][ng + 1] = v.y;
      Bs[kl][ng + 2] = v.z; Bs[kl][ng + 3] = v.w;
    }
    __syncthreads();
    for (int ks = 0; ks < 8; ++ks) {
      const int kb = ks * 4 + ((lane >> 4) << 1);
      v2f af;
      int mm = wm * 16 + (lane & 15);
      af.x = As[kb][mm]; af.y = As[kb + 1][mm];
      for (int ni = 0; ni < 2; ++ni) {
        int nn = wn * 32 + ni * 16 + (lane & 15);
        v2f bf; bf.x = Bs[kb][nn]; bf.y = Bs[kb + 1][nn];
        acc[ni] = wmma4(af, bf, acc[ni]);
      }
    }
    __syncthreads();
  }
  const int hiM = (lane >> 4) << 3;
  for (int ni = 0; ni < 2; ++ni)
    for (int r = 0; r < 8; ++r) {
      int c = wm * 16 + r + hiM;
      if (c < NCLS) {
        int n = nT0 + wn * 32 + ni * 16 + (lane & 15);
        wr[(((size_t)(b * NSC + s)) * NCLS + c) * NPIX + n] = acc[ni][r] + b1[c];
      }
    }
}

// =====================================================================
// K5a: per (b,n): argmax over scales, class softmax -> wei_cum, scale_idx
// =====================================================================
__global__ __launch_bounds__(256)
void route_kernel(const float* __restrict__ wr,
                  float* __restrict__ weicum_out,
                  float* __restrict__ sidx_out,
                  int* __restrict__ idx_ws) {
  int g = blockIdx.x * 256 + threadIdx.x;   // 0..4095
  int b = g >> 10, n = g & 1023;
  float sel[NCLS];
  int   sid[NCLS];
  float mx = -3.4e38f;
  for (int c = 0; c < NCLS; ++c) {
    float best = -3.4e38f; int bs = 0;
    for (int s = 0; s < NSC; ++s) {
      float v = wr[(((size_t)(b * NSC + s)) * NCLS + c) * NPIX + n];
      if (v > best) { best = v; bs = s; }
    }
    sel[c] = best; sid[c] = bs;
    mx = fmaxf(mx, best);
  }
  float sum = 0.0f;
  for (int c = 0; c < NCLS; ++c) { sel[c] = __expf(sel[c] - mx); sum += sel[c]; }
  float inv = 1.0f / sum;
  for (int c = 0; c < NCLS; ++c) {
    size_t o = ((size_t)b * NCLS + c) * NPIX + n;
    weicum_out[o] = sel[c] * inv;
    sidx_out[o]   = (float)sid[c];
    idx_ws[o]     = sid[c];
  }
}

// =====================================================================
// K5b: wei_pool[b,c,:] = softmax_n( wei_cum[b,c,:] * exp(wp_w[c]) )
// =====================================================================
__global__ __launch_bounds__(256)
void pool_kernel(const float* __restrict__ weicum,
                 const float* __restrict__ wpw,
                 float* __restrict__ weipool_out) {
  __shared__ float red[256];
  int bc = blockIdx.x;            // 0..83
  int c  = bc % NCLS;
  float temp = __expf(wpw[c]);
  const float* src = weicum + (size_t)bc * NPIX;
  float v[4];
  float mx = -3.4e38f;
  for (int j = 0; j < 4; ++j) {
    v[j] = src[threadIdx.x + j * 256] * temp;
    mx = fmaxf(mx, v[j]);
  }
  red[threadIdx.x] = mx; __syncthreads();
  for (int off = 128; off > 0; off >>= 1) {
    if (threadIdx.x < off) red[threadIdx.x] = fmaxf(red[threadIdx.x], red[threadIdx.x + off]);
    __syncthreads();
  }
  mx = red[0]; __syncthreads();
  float part = 0.0f;
  for (int j = 0; j < 4; ++j) { v[j] = __expf(v[j] - mx); part += v[j]; }
  red[threadIdx.x] = part; __syncthreads();
  for (int off = 128; off > 0; off >>= 1) {
    if (threadIdx.x < off) red[threadIdx.x] += red[threadIdx.x + off];
    __syncthreads();
  }
  float inv = 1.0f / red[0];
  float* dst = weipool_out + (size_t)bc * NPIX;
  for (int j = 0; j < 4; ++j) dst[threadIdx.x + j * 256] = v[j] * inv;
}

// =====================================================================
// K5c: wcomb[b,c,s,n] = (s == idx[b,c,n]) ? wei_pool[b,c,n] : 0
// =====================================================================
__global__ __launch_bounds__(256)
void wcomb_kernel(const int* __restrict__ idx_ws,
                  const float* __restrict__ weipool,
                  float* __restrict__ wcomb) {
  size_t g = (size_t)blockIdx.x * 256 + threadIdx.x;   // 0..86015
  int n = g & 1023;
  int bc = (int)(g >> 10);
  int ss = idx_ws[g];
  float wp = weipool[g];
  float* dst = wcomb + ((size_t)bc * NSC) * NPIX + n;
  for (int s = 0; s < NSC; ++s) dst[(size_t)s * NPIX] = (s == ss) ? wp : 0.0f;
}

// =====================================================================
// K6: feat_sg[b,c,d] = sum_{s,n} wcomb[b,c,s,n] * out_diff[b,s,d,n]
//     per b: M=32(21), N tile 64 (d), K=7168, Kt=32
// =====================================================================
__global__ __launch_bounds__(256)
void fsg_gemm_kernel(const float* __restrict__ wcomb,
                     const float* __restrict__ od,
                     float* __restrict__ fsg) {
  __shared__ float As[32][48];    // [k][c]
  __shared__ float Bs[32][80];    // [k][d]
  const int t = threadIdx.x, lane = t & 31, wid = t >> 5;
  const int wm = wid >> 2, wn = wid & 3;
  const int b = blockIdx.z;
  const int dT0 = blockIdx.x * 64;
  const float* A = wcomb + ((size_t)b * NCLS) * (NSC * NPIX);   // [c][s*1024+n]

  v8f acc = (v8f)0.0f;

  for (int kt = 0; kt < NSC * NPIX; kt += 32) {
    {
      int cl = t >> 3, kg = (t & 7) << 2;
      float4 v = make_float4(0.f, 0.f, 0.f, 0.f);
      if (cl < NCLS) v = *(const float4*)(A + (size_t)cl * (NSC * NPIX) + kt + kg);
      As[kg + 0][cl] = v.x; As[kg + 1][cl] = v.y;
      As[kg + 2][cl] = v.z; As[kg + 3][cl] = v.w;
    }
    {
      int s = kt >> 10;
      int n0 = kt & 1023;
      const float* B = od + (((size_t)b * NSC + s) << 10) * NPIX;
      for (int i = 0; i < 2; ++i) {
        int f4 = i * 256 + t;
        int dl = f4 >> 3, kg = (f4 & 7) << 2;
        const float4 v = *(const float4*)(B + (size_t)(dT0 + dl) * NPIX + n0 + kg);
        Bs[kg + 0][dl] = v.x; Bs[kg + 1][dl] = v.y;
        Bs[kg + 2][dl] = v.z; Bs[kg + 3][dl] = v.w;
      }
    }
    __syncthreads();
    for (int ks = 0; ks < 8; ++ks) {
      const int kb = ks * 4 + ((lane >> 4) << 1);
      v2f af, bf;
      int mm = wm * 16 + (lane & 15);
      int nn = wn * 16 + (lane & 15);
      af.x = As[kb][mm]; af.y = As[kb + 1][mm];
      bf.x = Bs[kb][nn]; bf.y = Bs[kb + 1][nn];
      acc = wmma4(af, bf, acc);
    }
    __syncthreads();
  }
  const int hiM = (lane >> 4) << 3;
  for (int r = 0; r < 8; ++r) {
    int c = wm * 16 + r + hiM;
    if (c < NCLS) {
      int d = dT0 + wn * 16 + (lane & 15);
      fsg[((size_t)b * NCLS + c) * DCH + d] = acc[r];
    }
  }
}

// =====================================================================
// K7: prob[b,c] = sigmoid( sum_d feat_sg*w1[c,d] + D*b1[c] )
// =====================================================================
__global__ __launch_bounds__(256)
void prob_kernel(const float* __restrict__ fsg,
                 const float* __restrict__ w1,
                 const float* __restrict__ b1,
                 float* __restrict__ prob_out) {
  __shared__ float red[256];
  int bc = blockIdx.x;
  int b = bc / NCLS, c = bc - b * NCLS;
  const float* f = fsg + (size_t)bc * DCH;
  const float* w = w1 + (size_t)c * DCH;
  float part = 0.0f;
  for (int j = 0; j < 4; ++j) {
    int d = threadIdx.x + j * 256;
    part += f[d] * w[d];
  }
  red[threadIdx.x] = part; __syncthreads();
  for (int off = 128; off > 0; off >>= 1) {
    if (threadIdx.x < off) red[threadIdx.x] += red[threadIdx.x + off];
    __syncthreads();
  }
  if (threadIdx.x == 0) {
    float v = red[0] + (float)DCH * b1[c];
    prob_out[(size_t)b * NCLS + c] = 1.0f / (1.0f + __expf(-v));
  }
}

// =====================================================================
// launch
// =====================================================================
extern "C" void kernel_launch(void* const* d_in, const int* in_sizes, int n_in,
                              void* d_out, int out_size, void* d_ws, size_t ws_size,
                              hipStream_t stream) {
  const float* x      = (const float*)d_in[0];
  const float* diffW  = (const float*)d_in[1];
  const float* conv_w = (const float*)d_in[2];
  const float* conv_b = (const float*)d_in[3];
  const float* w1     = (const float*)d_in[4];
  const float* b1     = (const float*)d_in[5];
  const float* wp_w   = (const float*)d_in[6];

  float* out = (float*)d_out;
  float* prob_o    = out;                         // 84
  float* weicum_o  = out + 84;                    // 86016
  float* weipool_o = out + 84 + 86016;            // 86016
  float* sidx_o    = out + 84 + 2 * 86016;        // 86016

  // workspace layout (floats)
  float* ws = (float*)d_ws;
  float* feat  = ws;                                          // 4*1024*1024
  float* dn    = feat  + (size_t)BSZ * DCH * NPIX;            // 7*4*1024*1024
  float* od    = dn    + (size_t)NSC * BSZ * NPIX * NPIX;     // 7*4*1024*1024
  float* wraw  = od    + (size_t)NSC * BSZ * DCH * NPIX;      // 4*7*21*1024
  int*   idxb  = (int*)(wraw + (size_t)BSZ * NSC * NCLS * NPIX);  // 4*21*1024 ints
  float* wcomb = (float*)(idxb + (size_t)BSZ * NCLS * NPIX);  // 4*21*7168
  float* fsg   = wcomb + (size_t)BSZ * NCLS * NSC * NPIX;     // 4*21*1024

  // 1) dilated conv (implicit GEMM, fp32 WMMA)
  conv_gemm_kernel<<<dim3(32, 8), 256, 0, stream>>>(x, conv_w, conv_b, feat);
  // 2) multiscale pow/threshold/normalize
  dn_kernel<<<dim3(NSC * BSZ * NPIX), 256, 0, stream>>>(diffW, dn);
  // 3) 28x diffusion GEMMs (async double-buffered staging)
  diff_gemm_kernel<<<dim3(8, 8, BSZ * NSC), 256, 0, stream>>>(feat, dn, od);
  // 4) class head GEMMs
  cls_gemm_kernel<<<dim3(8, 1, BSZ * NSC), 256, 0, stream>>>(od, w1, b1, wraw);
  // 5) routing + class softmax
  route_kernel<<<dim3(16), 256, 0, stream>>>(wraw, weicum_o, sidx_o, idxb);
  // 6) spatial temperature softmax
  pool_kernel<<<dim3(BSZ * NCLS), 256, 0, stream>>>(weicum_o, wp_w, weipool_o);
  // 7) dense routed pooling weights
  wcomb_kernel<<<dim3(336), 256, 0, stream>>>(idxb, weipool_o, wcomb);
  // 8) pooled-feature GEMM
  fsg_gemm_kernel<<<dim3(16, 1, BSZ), 256, 0, stream>>>(wcomb, od, fsg);
  // 9) sigmoid scores
  prob_kernel<<<dim3(BSZ * NCLS), 256, 0, stream>>>(fsg, w1, b1, prob_o);
}